// CrossMambaBlock_83150566850925
// MI455X (gfx1250) — compile-verified
//
#include <hip/hip_runtime.h>
#include <math.h>

typedef float v2f __attribute__((ext_vector_type(2)));
typedef float v8f __attribute__((ext_vector_type(8)));

#define DIM     128
#define LSEQ    4096
#define BSZ     4
#define BL      (BSZ*LSEQ)      // 16384 tokens
#define DSTATE  16
#define DTRANK  8
#define NPROJ   40              // DT_RANK + 2*D_STATE
#define LCHUNK  64
#define NCHUNK  (LSEQ/LCHUNK)   // 64
#define NJOB    (BSZ*2)         // 4 batches x 2 directions

enum { EPI_NONE = 0, EPI_SIGMOID = 1, EPI_SOFTPLUS = 2 };

__device__ __forceinline__ float sigmoidf_(float x) { return 1.f / (1.f + __expf(-x)); }
__device__ __forceinline__ float siluf_(float x)    { return x * sigmoidf_(x); }
__device__ __forceinline__ float softplusf_(float x){ return (x > 20.f) ? x : __logf(1.f + __expf(x)); }

// ---------------------------------------------------------------------------
// LayerNorm of input0 and input1 (one wave32 per token, 4 channels per lane)
// ---------------------------------------------------------------------------
__global__ __launch_bounds__(256) void ln2_kernel(
    const float* __restrict__ in0, const float* __restrict__ in1,
    const float* __restrict__ w0, const float* __restrict__ b0,
    const float* __restrict__ w1, const float* __restrict__ b1,
    float* __restrict__ o0, float* __restrict__ o1)
{
    int wave = threadIdx.x >> 5;
    int lane = threadIdx.x & 31;
    int tok  = blockIdx.x * 8 + wave;
    if (tok >= BL) return;
    const float* p0 = in0 + (size_t)tok * DIM;
    const float* p1 = in1 + (size_t)tok * DIM;
    float v0[4], v1[4], s0 = 0.f, s1 = 0.f, q0 = 0.f, q1 = 0.f;
#pragma unroll
    for (int i = 0; i < 4; i++) {
        v0[i] = p0[lane + 32 * i]; v1[i] = p1[lane + 32 * i];
        s0 += v0[i]; q0 += v0[i] * v0[i];
        s1 += v1[i]; q1 += v1[i] * v1[i];
    }
#pragma unroll
    for (int m = 16; m >= 1; m >>= 1) {
        s0 += __shfl_xor(s0, m); q0 += __shfl_xor(q0, m);
        s1 += __shfl_xor(s1, m); q1 += __shfl_xor(q1, m);
    }
    float mu0 = s0 * (1.f / DIM), mu1 = s1 * (1.f / DIM);
    float r0 = rsqrtf(q0 * (1.f / DIM) - mu0 * mu0 + 1e-5f);
    float r1 = rsqrtf(q1 * (1.f / DIM) - mu1 * mu1 + 1e-5f);
#pragma unroll
    for (int i = 0; i < 4; i++) {
        int c = lane + 32 * i;
        o0[(size_t)tok * DIM + c] = (v0[i] - mu0) * r0 * w0[c] + b0[c];
        o1[(size_t)tok * DIM + c] = (v1[i] - mu1) * r1 * w1[c] + b1[c];
    }
}

// ---------------------------------------------------------------------------
// Generic fp32 WMMA GEMM:  C[M,N] = epi( [A0|A1][M, K0+K1] @ W[K,N] + bias )
// Block = 256 threads = 8 waves; each wave owns one 16x16 tile.
// A tile staged through LDS, K-loop in steps of 4 via v_wmma_f32_16x16x4_f32.
// For N not a multiple of 16 the out-of-range lanes load a clamped column
// (their results land in padded output columns, discarded at the store) --
// this keeps the inner loop branch-free and EXEC all-ones for WMMA.
// ---------------------------------------------------------------------------
__global__ __launch_bounds__(256) void wmma_gemm_kernel(
    const float* __restrict__ A0, int lda0, int K0,
    const float* __restrict__ A1, int lda1, int K1,
    const float* __restrict__ W,  int ldw,
    const float* __restrict__ bias,
    float* __restrict__ C, int ldc, int N, int epi)
{
    __shared__ float ldsA[16 * 256];
    const int Kt    = K0 + K1;
    const int mbase = blockIdx.x * 16;

    for (int idx = threadIdx.x; idx < 16 * Kt; idx += 256) {
        int row = idx / Kt, k = idx - row * Kt;
        float v = (k < K0) ? A0[(size_t)(mbase + row) * lda0 + k]
                           : A1[(size_t)(mbase + row) * lda1 + (k - K0)];
        ldsA[row * Kt + k] = v;
    }
    __syncthreads();

    const int wave = threadIdx.x >> 5;
    const int lane = threadIdx.x & 31;
    const int half = lane >> 4;        // lanes 16-31 carry K+2 / M+8 halves
    const int r    = lane & 15;
    const int ntile  = blockIdx.y * 8 + wave;
    const int ntiles = (N + 15) >> 4;
    if (ntile >= ntiles) return;       // whole-wave exit: EXEC stays all-ones for WMMA
    const int nbase = ntile * 16;
    const int col   = nbase + r;
    const bool colok = (col < N);
    const int colL  = colok ? col : (N - 1);   // clamped, branch-free load column
    const float* __restrict__ Wl = W + colL;

    v8f acc = {0.f, 0.f, 0.f, 0.f, 0.f, 0.f, 0.f, 0.f};
    for (int k0 = 0; k0 < Kt; k0 += 4) {
        v2f a, b;
        int ka = k0 + 2 * half;
        a[0] = ldsA[r * Kt + ka];
        a[1] = ldsA[r * Kt + ka + 1];
        b[0] = Wl[(size_t)ka * ldw];
        b[1] = Wl[(size_t)(ka + 1) * ldw];
        acc = __builtin_amdgcn_wmma_f32_16x16x4_f32(false, a, false, b,
                                                    (short)0, acc, false, false);
    }
    if (!colok) return;
    float bv = bias ? bias[col] : 0.f;
#pragma unroll
    for (int v = 0; v < 8; v++) {
        int row = mbase + v + 8 * half;
        float x = acc[v] + bv;
        if (epi == EPI_SIGMOID)       x = sigmoidf_(x);
        else if (epi == EPI_SOFTPLUS) x = softplusf_(x);
        C[(size_t)row * ldc + col] = x;
    }
}

// ---------------------------------------------------------------------------
// Depthwise causal conv (width 4) + SiLU for both x (inside xz) and xe.
// ---------------------------------------------------------------------------
__global__ __launch_bounds__(256) void conv_silu_kernel(
    const float* __restrict__ xz, const float* __restrict__ xe,
    const float* __restrict__ cw, const float* __restrict__ cb,
    const float* __restrict__ cwe, const float* __restrict__ cbe,
    float* __restrict__ xo, float* __restrict__ xeo)
{
    int idx = blockIdx.x * blockDim.x + threadIdx.x;
    if (idx >= BL * DIM) return;
    int c   = idx & (DIM - 1);
    int tok = idx >> 7;
    int pos = tok & (LSEQ - 1);
    float acc1 = cb[c], acc2 = cbe[c];
#pragma unroll
    for (int k = 0; k < 4; k++) {
        int sp = pos - 3 + k;
        if (sp >= 0) {
            int st = tok - 3 + k;
            acc1 += xz[(size_t)st * 256 + c] * cw[c * 4 + k];
            acc2 += xe[(size_t)st * DIM + c] * cwe[c * 4 + k];
        }
    }
    xo[idx]  = siluf_(acc1);
    xeo[idx] = siluf_(acc2);
}

// ---------------------------------------------------------------------------
// Selective scan, chunked 3-pass. h_t = exp(dt*A)*h + dt*u*B ; y = <h,C> + u*D
// Pass 1: per (job,chunk) block (128 thr = 1 channel/thr, 16 states in regs)
//         compute aprod = prod(dA) and local h with h0=0.
// ---------------------------------------------------------------------------
__global__ __launch_bounds__(128) void scan_pass1_kernel(
    const float* __restrict__ dt, const float* __restrict__ u,
    const float* __restrict__ dtBC,
    const float* __restrict__ A_log, const float* __restrict__ A_log_rev,
    float* __restrict__ aprod, float* __restrict__ hloc)
{
    __shared__ float sB[8 * 16];
    int chunk = blockIdx.x, j = blockIdx.y;
    int b = j >> 1, dir = j & 1;
    int c = threadIdx.x;
    const float* Aln = dir ? A_log_rev : A_log;
    float Ac[DSTATE];
#pragma unroll
    for (int s = 0; s < DSTATE; s++) Ac[s] = -__expf(Aln[c * DSTATE + s]);
    float h[DSTATE], a[DSTATE];
#pragma unroll
    for (int s = 0; s < DSTATE; s++) { h[s] = 0.f; a[s] = 1.f; }

    for (int t8 = 0; t8 < 8; t8++) {
        {   // stage B vectors for the next 8 steps into LDS (coalesced)
            int sl = threadIdx.x >> 4;        // step 0..7
            int s  = threadIdx.x & 15;
            int tg  = chunk * LCHUNK + t8 * 8 + sl;
            int pos = dir ? (LSEQ - 1 - tg) : tg;
            int tok = b * LSEQ + pos;
            __syncthreads();
            sB[threadIdx.x] = dtBC[(size_t)tok * NPROJ + DTRANK + s];
            __syncthreads();
        }
        for (int tt = 0; tt < 8; tt++) {
            int tg  = chunk * LCHUNK + t8 * 8 + tt;
            int pos = dir ? (LSEQ - 1 - tg) : tg;
            int tok = b * LSEQ + pos;
            float dtv = dt[(size_t)tok * DIM + c];
            float du  = dtv * u[(size_t)tok * DIM + c];
#pragma unroll
            for (int s = 0; s < DSTATE; s++) {
                float dA = __expf(dtv * Ac[s]);
                h[s] = dA * h[s] + du * sB[tt * 16 + s];
                a[s] = a[s] * dA;
            }
        }
    }
    size_t base = ((size_t)(j * NCHUNK + chunk) * DIM + c) * DSTATE;
#pragma unroll
    for (int s = 0; s < DSTATE; s++) { aprod[base + s] = a[s]; hloc[base + s] = h[s]; }
}

// Pass 2: serial scan over 64 chunk summaries; 16384 independent state lanes.
__global__ __launch_bounds__(256) void scan_pass2_kernel(
    const float* __restrict__ aprod, const float* __restrict__ hloc,
    float* __restrict__ h0buf)
{
    int gid = blockIdx.x * blockDim.x + threadIdx.x;
    if (gid >= NJOB * DIM * DSTATE) return;
    int j = gid / (DIM * DSTATE);
    int p = gid - j * (DIM * DSTATE);
    float h = 0.f;
    for (int g = 0; g < NCHUNK; g++) {
        size_t idx = (size_t)(j * NCHUNK + g) * (DIM * DSTATE) + p;
        h0buf[idx] = h;
        h = aprod[idx] * h + hloc[idx];
    }
}

// Pass 3: replay each chunk with correct h0, emit y (+ u*D). B,C staged in LDS.
__global__ __launch_bounds__(128) void scan_pass3_kernel(
    const float* __restrict__ dt, const float* __restrict__ u,
    const float* __restrict__ dtBC,
    const float* __restrict__ A_log, const float* __restrict__ A_log_rev,
    const float* __restrict__ Dp, const float* __restrict__ Dp_rev,
    const float* __restrict__ h0buf,
    float* __restrict__ yf, float* __restrict__ yb)
{
    __shared__ float sBC[4 * 32];
    int chunk = blockIdx.x, j = blockIdx.y;
    int b = j >> 1, dir = j & 1;
    int c = threadIdx.x;
    const float* Aln = dir ? A_log_rev : A_log;
    float Dc  = dir ? Dp_rev[c] : Dp[c];
    float* yo = dir ? yb : yf;
    float Ac[DSTATE];
#pragma unroll
    for (int s = 0; s < DSTATE; s++) Ac[s] = -__expf(Aln[c * DSTATE + s]);
    float h[DSTATE];
    size_t hbase = (size_t)(j * NCHUNK + chunk) * (DIM * DSTATE) + c * DSTATE;
#pragma unroll
    for (int s = 0; s < DSTATE; s++) h[s] = h0buf[hbase + s];

    for (int t4 = 0; t4 < 16; t4++) {
        {   // stage B+C (32 floats) for the next 4 steps
            int sl = threadIdx.x >> 5;        // step 0..3
            int q  = threadIdx.x & 31;
            int tg  = chunk * LCHUNK + t4 * 4 + sl;
            int pos = dir ? (LSEQ - 1 - tg) : tg;
            int tok = b * LSEQ + pos;
            __syncthreads();
            sBC[threadIdx.x] = dtBC[(size_t)tok * NPROJ + DTRANK + q];
            __syncthreads();
        }
        for (int tt = 0; tt < 4; tt++) {
            int tg  = chunk * LCHUNK + t4 * 4 + tt;
            int pos = dir ? (LSEQ - 1 - tg) : tg;
            int tok = b * LSEQ + pos;
            float dtv = dt[(size_t)tok * DIM + c];
            float uv  = u[(size_t)tok * DIM + c];
            float du  = dtv * uv;
            float y = 0.f;
#pragma unroll
            for (int s = 0; s < DSTATE; s++) {
                float dA = __expf(dtv * Ac[s]);
                h[s] = dA * h[s] + du * sBC[tt * 32 + s];
                y += h[s] * sBC[tt * 32 + 16 + s];
            }
            yo[(size_t)tok * DIM + c] = y + uv * Dc;   // bwd writes at flipped tok == un-flipped output
        }
    }
}

// ---------------------------------------------------------------------------
// Tail: gate + RMSNorm + out_proj (WMMA) + weighted residual mix + skip.
// ---------------------------------------------------------------------------
__global__ __launch_bounds__(256) void final_kernel(
    const float* __restrict__ yf, const float* __restrict__ yb,
    const float* __restrict__ xz,
    const float* __restrict__ rms_w, const float* __restrict__ Wout,
    const float* __restrict__ weight, const float* __restrict__ x0n,
    const float* __restrict__ in0, float* __restrict__ out)
{
    __shared__ float yn[16 * DIM];
    const int mbase = blockIdx.x * 16;
    const int t   = threadIdx.x >> 4;   // token in tile (16 lanes per token)
    const int sub = threadIdx.x & 15;
    int tok = mbase + t;
    float ss = 0.f, vloc[8];
#pragma unroll
    for (int i = 0; i < 8; i++) {
        int c = sub * 8 + i;
        size_t g = (size_t)tok * DIM + c;
        float z = xz[(size_t)tok * 256 + DIM + c];
        float v = (yf[g] + yb[g]) * 0.5f * siluf_(z);
        vloc[i] = v;
        ss += v * v;
    }
#pragma unroll
    for (int m = 8; m >= 1; m >>= 1) ss += __shfl_xor(ss, m);   // 16-lane butterfly
    float scale = rsqrtf(ss * (1.f / DIM) + 1e-5f);
#pragma unroll
    for (int i = 0; i < 8; i++) {
        int c = sub * 8 + i;
        yn[t * DIM + c] = vloc[i] * scale * rms_w[c];
    }
    __syncthreads();

    const int wave = threadIdx.x >> 5;
    const int lane = threadIdx.x & 31;
    const int half = lane >> 4;
    const int r    = lane & 15;
    const int col  = wave * 16 + r;
    v8f acc = {0.f, 0.f, 0.f, 0.f, 0.f, 0.f, 0.f, 0.f};
    for (int k0 = 0; k0 < DIM; k0 += 4) {
        v2f a, b;
        int ka = k0 + 2 * half;
        a[0] = yn[r * DIM + ka];
        a[1] = yn[r * DIM + ka + 1];
        b[0] = Wout[(size_t)ka * DIM + col];
        b[1] = Wout[(size_t)(ka + 1) * DIM + col];
        acc = __builtin_amdgcn_wmma_f32_16x16x4_f32(false, a, false, b,
                                                    (short)0, acc, false, false);
    }
#pragma unroll
    for (int v = 0; v < 8; v++) {
        int row = mbase + v + 8 * half;
        size_t g = (size_t)row * DIM + col;
        float wv = weight[g];
        out[g] = acc[v] * wv + x0n[g] * (1.f - wv) + in0[g];
    }
}

// ---------------------------------------------------------------------------
extern "C" void kernel_launch(void* const* d_in, const int* in_sizes, int n_in,
                              void* d_out, int out_size, void* d_ws, size_t ws_size,
                              hipStream_t stream)
{
    (void)in_sizes; (void)n_in; (void)out_size; (void)ws_size;
    const float* input0          = (const float*)d_in[0];
    const float* input1          = (const float*)d_in[1];
    const float* ln0_w           = (const float*)d_in[2];
    const float* ln0_b           = (const float*)d_in[3];
    const float* ln1_w           = (const float*)d_in[4];
    const float* ln1_b           = (const float*)d_in[5];
    const float* cw_w            = (const float*)d_in[6];
    const float* cw_b            = (const float*)d_in[7];
    const float* in_proj_w       = (const float*)d_in[8];
    const float* in_proj_extra_w = (const float*)d_in[9];
    const float* conv_w          = (const float*)d_in[10];
    const float* conv_b          = (const float*)d_in[11];
    const float* conv_we         = (const float*)d_in[12];
    const float* conv_be         = (const float*)d_in[13];
    const float* x_proj_w        = (const float*)d_in[14];
    const float* dt_proj_w       = (const float*)d_in[15];
    const float* dt_proj_b       = (const float*)d_in[16];
    const float* A_log           = (const float*)d_in[17];
    const float* A_log_rev       = (const float*)d_in[18];
    const float* Dvec            = (const float*)d_in[19];
    const float* D_rev           = (const float*)d_in[20];
    const float* rms_w           = (const float*)d_in[21];
    const float* out_proj_w      = (const float*)d_in[22];

    float* ws = (float*)d_ws;
    size_t off = 0;
    auto alloc = [&](size_t n) { float* p = ws + off; off += n; return p; };
    float* x0n     = alloc((size_t)BL * DIM);
    float* x1n     = alloc((size_t)BL * DIM);     // dead after xe gemm -> reused as y_b
    float* weightB = alloc((size_t)BL * DIM);
    float* xz      = alloc((size_t)BL * 256);
    float* xe_pre  = alloc((size_t)BL * DIM);     // dead after conv -> reused as y_f
    float* x_conv  = alloc((size_t)BL * DIM);
    float* xe_conv = alloc((size_t)BL * DIM);
    float* dtBC    = alloc((size_t)BL * NPROJ);
    float* dtb     = alloc((size_t)BL * DIM);
    float* aprod   = alloc((size_t)NJOB * NCHUNK * DIM * DSTATE);
    float* hloc    = alloc((size_t)NJOB * NCHUNK * DIM * DSTATE);
    float* h0buf   = alloc((size_t)NJOB * NCHUNK * DIM * DSTATE);
    float* yf = xe_pre;
    float* yb = x1n;

    ln2_kernel<<<BL / 8, 256, 0, stream>>>(input0, input1, ln0_w, ln0_b, ln1_w, ln1_b, x0n, x1n);

    // weight = sigmoid([x0n|x1n] @ cw_w + cw_b)        K=256, N=128
    wmma_gemm_kernel<<<dim3(BL / 16, 1), 256, 0, stream>>>(
        x0n, DIM, DIM, x1n, DIM, DIM, cw_w, DIM, cw_b, weightB, DIM, DIM, EPI_SIGMOID);
    // xz = x0n @ in_proj_w                             K=128, N=256
    wmma_gemm_kernel<<<dim3(BL / 16, 2), 256, 0, stream>>>(
        x0n, DIM, DIM, nullptr, 0, 0, in_proj_w, 256, nullptr, xz, 256, 256, EPI_NONE);
    // xe = x1n @ in_proj_extra_w                       K=128, N=128
    wmma_gemm_kernel<<<dim3(BL / 16, 1), 256, 0, stream>>>(
        x1n, DIM, DIM, nullptr, 0, 0, in_proj_extra_w, DIM, nullptr, xe_pre, DIM, DIM, EPI_NONE);

    conv_silu_kernel<<<(BL * DIM) / 256, 256, 0, stream>>>(
        xz, xe_pre, conv_w, conv_b, conv_we, conv_be, x_conv, xe_conv);

    // proj = xe_conv @ x_proj_w                        K=128, N=40 (padded to 48)
    wmma_gemm_kernel<<<dim3(BL / 16, 1), 256, 0, stream>>>(
        xe_conv, DIM, DIM, nullptr, 0, 0, x_proj_w, NPROJ, nullptr, dtBC, NPROJ, NPROJ, EPI_NONE);
    // dt = softplus(proj[:, :8] @ dt_proj_w + b)       K=8, N=128
    wmma_gemm_kernel<<<dim3(BL / 16, 1), 256, 0, stream>>>(
        dtBC, NPROJ, DTRANK, nullptr, 0, 0, dt_proj_w, DIM, dt_proj_b, dtb, DIM, DIM, EPI_SOFTPLUS);

    scan_pass1_kernel<<<dim3(NCHUNK, NJOB), 128, 0, stream>>>(
        dtb, x_conv, dtBC, A_log, A_log_rev, aprod, hloc);
    scan_pass2_kernel<<<(NJOB * DIM * DSTATE) / 256, 256, 0, stream>>>(aprod, hloc, h0buf);
    scan_pass3_kernel<<<dim3(NCHUNK, NJOB), 128, 0, stream>>>(
        dtb, x_conv, dtBC, A_log, A_log_rev, Dvec, D_rev, h0buf, yf, yb);

    final_kernel<<<BL / 16, 256, 0, stream>>>(
        yf, yb, xz, rms_w, out_proj_w, weightB, x0n, input0, (float*)d_out);
}